// EuclideanCodebook_86431921864876
// MI455X (gfx1250) — compile-verified
//
#include <hip/hip_runtime.h>

// ---------------------------------------------------------------------------
// VQ-VAE EuclideanCodebook forward for MI455X (gfx1250, wave32).
//
// dist GEMM: [M=16384, D=64] x [K=4096, D=64]^T in fp32 via
// V_WMMA_F32_16X16X4_F32 (16 chained WMMAs per 16x16 tile), fused with
// row-argmax, corrected to -(x^2 + e^2 - 2ab).
// Embed tiles are double-buffered in LDS via the CDNA5 async pipe
// (GLOBAL_LOAD_ASYNC_TO_LDS_B128 / ASYNCcnt), so the global->LDS copy of
// tile kb+1 overlaps the 16 WMMAs of tile kb.
// K-loop uses raw split barriers (s_barrier_signal/wait -1) instead of
// __syncthreads() so the non-temporal dist stores are NOT fenced per
// iteration (no s_wait_storecnt 0 on the critical path); LDS ordering is
// guaranteed by s_wait_asynccnt + the WMMA-consumed ds loads.
// dist (268 MB) is stream-written with non-temporal stores (> L2 = 192 MB,
// write-once). HBM floor ~11.5us @ 23.3 TB/s; fp32 WMMA compute is the
// same order, so the fused kernel sits near the roofline.
// ---------------------------------------------------------------------------

typedef __attribute__((ext_vector_type(2))) float v2f;
typedef __attribute__((ext_vector_type(8))) float v8f;

#define M_TOT 16384
#define DIMX  64
#define KCB   4096
#define DECAY 0.1f
#define LROW  68  // LDS row stride (floats): 16B-aligned rows, conflict-free

// Workgroup split barrier without the implicit global-store fence of
// __syncthreads(). "memory" clobber keeps LDS accesses ordered against it.
#define WG_BARRIER() \
  asm volatile("s_barrier_signal -1\n\ts_barrier_wait -1" ::: "memory")

// ---------------------------------------------------------------------------
// e_sq[k] = sum_d embed[k][d]^2
__global__ __launch_bounds__(64) void vq_esq(const float* __restrict__ embed,
                                             float* __restrict__ esq) {
  int k = blockIdx.x * blockDim.x + threadIdx.x;
  if (k >= KCB) return;
  const float* row = embed + k * DIMX;
  float s = 0.f;
#pragma unroll
  for (int d = 0; d < DIMX; ++d) s += row[d] * row[d];
  esq[k] = s;
}

__global__ void vq_zero(float* __restrict__ p, int n) {
  int i = blockIdx.x * blockDim.x + threadIdx.x;
  if (i < n) p[i] = 0.f;
}

// ---------------------------------------------------------------------------
// Async-stage one embed tile [16 cols][64 D] into an LDS buffer.
// 256 x 16B chunks; each of the 128 threads issues two async B128 copies.
__device__ __forceinline__ void issue_tile_async(
    const float* __restrict__ embed, float* lds_buf, int kb, int t) {
#pragma unroll
  for (int j = 0; j < 2; ++j) {
    int chunk = t * 2 + j;        // 0..255
    int r = chunk >> 4;           // tile column (embed row) 0..15
    int c0 = (chunk & 15) * 4;    // D offset in floats, 16B-aligned
    const float* g = embed + (size_t)(kb * 16 + r) * DIMX + c0;
    // Low 32 bits of the flat shared pointer are the LDS byte address.
    unsigned lds_addr = (unsigned)(unsigned long long)(&lds_buf[r * LROW + c0]);
    asm volatile("global_load_async_to_lds_b128 %0, %1, off"
                 :: "v"(lds_addr), "v"(g)
                 : "memory");
  }
}

// ---------------------------------------------------------------------------
// Main fused kernel: dist = 2*x.e^T - x^2 - e^2, row argmax, embed_ind.
// Block = 128 threads (4 waves); each wave owns 16 rows; block owns 64 rows.
// All waves walk the K dimension in 16-col tiles, double-buffered in LDS.
__global__ __launch_bounds__(128) void vq_dist_argmax(
    const float* __restrict__ x, const float* __restrict__ embed,
    const float* __restrict__ esq, float* __restrict__ dist,
    float* __restrict__ embed_ind, int* __restrict__ idx_out) {
  __shared__ float lds_tile[2][16 * LROW];  // 2 x 4.25KB double buffer

  const int lane = threadIdx.x & 31;
  const int col_in_tile = lane & 15;   // A row index / B-C-D column index
  const int hi = lane >> 4;            // 0: k%4 in {0,1}; 1: k%4 in {2,3}
  const int doff = hi * 2;
  const int wave = threadIdx.x >> 5;
  const int row_base = blockIdx.x * 64 + wave * 16;

  // --- A fragments: 16 rows x 64 D, resident in VGPRs for the whole K loop.
  // ISA 32-bit A 16x4 layout: lanes 0-15 hold {k,k+1}, lanes 16-31 {k+2,k+3}.
  const float* xrow = x + (size_t)(row_base + col_in_tile) * DIMX;
  v2f a[16];
  float part = 0.f;
#pragma unroll
  for (int c = 0; c < 16; ++c) {
    a[c] = *(const v2f*)(xrow + 4 * c + doff);
    part += a[c].x * a[c].x + a[c].y * a[c].y;
  }
  // Lane holds half of its row's |x|^2; partner lane (^16) holds the rest.
  float xsq_full = part + __shfl_xor(part, 16, 32);
  // Redistribute to match C/D layout: acc VGPR r = row r (lanes<16) / r+8.
  float xr[8];
#pragma unroll
  for (int r = 0; r < 8; ++r) xr[r] = __shfl(xsq_full, r + hi * 8, 32);

  float bestv[8];
  int besti[8];
#pragma unroll
  for (int r = 0; r < 8; ++r) { bestv[r] = -3.4e38f; besti[r] = 0; }

  // Prologue: async-stage tile 0.
  issue_tile_async(embed, lds_tile[0], 0, threadIdx.x);

  for (int kb = 0; kb < KCB / 16; ++kb) {
    float* cur = lds_tile[kb & 1];

    // Issue next tile into the other buffer, then wait for the current
    // tile's own async loads (in-order per wave: <=2 outstanding suffices).
    if (kb + 1 < KCB / 16) {
      issue_tile_async(embed, lds_tile[(kb + 1) & 1], kb + 1, threadIdx.x);
      asm volatile("s_wait_asynccnt 0x2" ::: "memory");
    } else {
      asm volatile("s_wait_asynccnt 0x0" ::: "memory");
    }
    WG_BARRIER();  // publish all lanes' async LDS writes for tile kb

    // --- B fragments from LDS (4x16 per step; same k-split as A).
    v2f b[16];
#pragma unroll
    for (int c = 0; c < 16; ++c)
      b[c] = *(const v2f*)&cur[col_in_tile * LROW + 4 * c + doff];

    // --- 16 chained fp32 WMMAs: acc(16x16) += A(16x4) x B(4x16).
    v8f acc = {};
#pragma unroll
    for (int c = 0; c < 16; ++c)
      acc = __builtin_amdgcn_wmma_f32_16x16x4_f32(
          false, a[c], false, b[c], (short)0, acc, false, false);

    const int col = kb * 16 + col_in_tile;
    const float ecol = esq[col];
#pragma unroll
    for (int r = 0; r < 8; ++r) {
      float dval = 2.f * acc[r] - xr[r] - ecol;  // -(x^2 + e^2 - 2ab)
      int mrow = row_base + r + hi * 8;
      __builtin_nontemporal_store(dval, dist + (size_t)mrow * KCB + col);
      if (dval > bestv[r]) { bestv[r] = dval; besti[r] = col; }
    }
    // All lanes' B-fragment ds loads are complete (consumed by the WMMAs),
    // so after this barrier tile kb's buffer may be re-issued. NT dist
    // stores intentionally NOT fenced here — they drain in the background.
    WG_BARRIER();
  }

  // --- Argmax across the 16 lanes holding one row (xor masks < 16 stay in
  // each wave32 half). Tie-break: smallest index, matching jnp.argmax.
#pragma unroll
  for (int r = 0; r < 8; ++r) {
#pragma unroll
    for (int off = 8; off >= 1; off >>= 1) {
      float ov = __shfl_xor(bestv[r], off, 32);
      int oi = __shfl_xor(besti[r], off, 32);
      if (ov > bestv[r] || (ov == bestv[r] && oi < besti[r])) {
        bestv[r] = ov;
        besti[r] = oi;
      }
    }
    if (col_in_tile == 0) {
      int mrow = row_base + r + hi * 8;
      idx_out[mrow] = besti[r];
      embed_ind[mrow] = (float)besti[r];
    }
  }
}

// ---------------------------------------------------------------------------
// quantize gather + EMA scatter-add (bins, embed_sum) via fp32 global atomics.
__global__ __launch_bounds__(64) void vq_scatter(
    const float* __restrict__ x, const float* __restrict__ embed,
    const int* __restrict__ idx, float* __restrict__ quant,
    float* __restrict__ bins, float* __restrict__ esum) {
  int m = blockIdx.x;
  int d = threadIdx.x;
  int k = idx[m];
  quant[(size_t)m * DIMX + d] = embed[(size_t)k * DIMX + d];
  atomicAdd(&esum[(size_t)k * DIMX + d], x[(size_t)m * DIMX + d]);
  if (d == 0) atomicAdd(&bins[k], 1.0f);
}

// ---------------------------------------------------------------------------
// EMA blends: cluster_size_new, embed_avg_new.
__global__ void vq_finalize(const float* __restrict__ cs,
                            const float* __restrict__ ea,
                            const float* __restrict__ bins,
                            const float* __restrict__ esum,
                            float* __restrict__ cs_out,
                            float* __restrict__ ea_out) {
  int i = blockIdx.x * blockDim.x + threadIdx.x;
  if (i < KCB) cs_out[i] = cs[i] * DECAY + bins[i] * (1.f - DECAY);
  if (i < KCB * DIMX) ea_out[i] = ea[i] * DECAY + esum[i] * (1.f - DECAY);
}

// ---------------------------------------------------------------------------
extern "C" void kernel_launch(void* const* d_in, const int* in_sizes, int n_in,
                              void* d_out, int out_size, void* d_ws,
                              size_t ws_size, hipStream_t stream) {
  const float* x = (const float*)d_in[0];      // [8,2048,64]
  const float* embed = (const float*)d_in[1];  // [1,4096,64]
  const float* cs = (const float*)d_in[2];     // [1,4096]
  const float* ea = (const float*)d_in[3];     // [1,4096,64]

  // Outputs concatenated flat in return order.
  float* out = (float*)d_out;
  float* quant = out;                                   // 16384*64
  float* embed_ind = quant + (size_t)M_TOT * DIMX;      // 16384
  float* dist = embed_ind + M_TOT;                      // 16384*4096
  float* cs_out = dist + (size_t)M_TOT * KCB;           // 4096
  float* ea_out = cs_out + KCB;                         // 4096*64

  // Workspace (floats): [0,16384) idx(int) | [16384,20480) e_sq |
  // [20480,24576) bins | [24576,286720) embed_sum  (~1.1 MB)
  int* ws_idx = (int*)d_ws;
  float* wsf = (float*)d_ws;
  float* esq = wsf + M_TOT;
  float* bins = esq + KCB;
  float* esum = bins + KCB;

  // bins + esum are contiguous: zero both in one pass.
  int nzero = KCB + KCB * DIMX;
  vq_zero<<<(nzero + 255) / 256, 256, 0, stream>>>(bins, nzero);
  vq_esq<<<KCB / 64, 64, 0, stream>>>(embed, esq);
  vq_dist_argmax<<<M_TOT / 64, 128, 0, stream>>>(x, embed, esq, dist,
                                                 embed_ind, ws_idx);
  vq_scatter<<<M_TOT, 64, 0, stream>>>(x, embed, ws_idx, quant, bins, esum);
  vq_finalize<<<(KCB * DIMX + 255) / 256, 256, 0, stream>>>(cs, ea, bins, esum,
                                                            cs_out, ea_out);
}